// BilinearResizer_3599182594855
// MI455X (gfx1250) — compile-verified
//
#include <hip/hip_runtime.h>
#include <cstdint>

// Bilinear 2x upsample, (16,3,512,512) f32 -> (16,3,1024,1024) f32.
// Memory-bound (252 MB traffic, ~0.45 GFLOP -> ~10.8 us at 23.3 TB/s).
// CDNA5 paths used:
//   * GLOBAL_LOAD_ASYNC_TO_LDS_B128 (ASYNCcnt) to bulk-stage input rows in LDS
//   * s_wait_asynccnt + workgroup barrier for completion
//   * non-temporal B128 output stores (output > 192MB L2; keep 50MB input
//     resident in L2 for the overlapping neighbor reads)

#define HI 512
#define WI 512
#define WO 1024
#define TIR 8            // input rows consumed per block (-> 16 output rows)
#define PLANES 48        // 16 * 3

// Native clang vector type (HIP's float4 is a struct and is rejected by
// __builtin_nontemporal_store).
typedef float v4f __attribute__((ext_vector_type(4)));

__global__ __launch_bounds__(256)
void resize2x_async_kernel(const float* __restrict__ in, float* __restrict__ out) {
    // 9 rows: TIR rows + 1 bottom-neighbor row (clamped at image edge)
    __shared__ float tile[(TIR + 1) * WI];   // 4608 floats = 18 KB LDS

    const int plane = blockIdx.z;            // 0..47
    const int r0    = blockIdx.y * TIR;      // input row base for this tile
    const int tid   = threadIdx.x;           // 0..255

    const float* inP  = in  + (size_t)plane * HI * WI;
    float*       outP = out + (size_t)plane * (size_t)WO * WO;

    // ---- Stage 9 input rows into LDS via async global->LDS B128 copies ----
    const int TOTAL_V4 = (TIR + 1) * WI / 4;     // 1152 float4 chunks
    for (int base = 0; base < TOTAL_V4; base += 256) {
        int v = base + tid;
        if (v < TOTAL_V4) {
            int r   = v / (WI / 4);              // 0..8 (row within tile)
            int cv4 = v % (WI / 4);              // float4 column
            int rg  = min(r0 + r, HI - 1);       // clamp bottom edge
            const float* gsrc = inP + (size_t)rg * WI + (size_t)cv4 * 4;
            unsigned lds_off =
                (unsigned)(uintptr_t)(&tile[r * WI + cv4 * 4]); // LDS byte addr
            unsigned long long gaddr = (unsigned long long)(uintptr_t)gsrc;
            asm volatile("global_load_async_to_lds_b128 %0, %1, off"
                         :: "v"(lds_off), "v"(gaddr)
                         : "memory");
        }
    }
    // Wait for this wave's async copies, then rendezvous with the other waves.
    asm volatile("s_wait_asynccnt 0" ::: "memory");
    __syncthreads();

    // ---- Stencil: each thread emits a 2-row x 4-col output block per iter ----
    // tid indexes the column group: 2 input cols -> 4 output cols.
    // LDS access pattern: 32 lanes * stride-2 floats -> 32 distinct banks of 64
    // for the even-index reads and 32 distinct odd banks for the +1 reads: no
    // bank conflicts.
    const int cg = tid;                 // 0..255
    const int j0 = cg * 2;              // input column base
    const int j1 = j0 + 1;
    const int j2 = min(j0 + 2, WI - 1); // right-edge clamp

    #pragma unroll
    for (int ri = 0; ri < TIR; ++ri) {
        const float* row0 = &tile[ri * WI];
        const float* row1 = &tile[(ri + 1) * WI];

        float a = row0[j0], b = row0[j1], c = row0[j2];
        float d = row1[j0], e = row1[j1], f = row1[j2];

        v4f top = { a,
                    0.5f * (a + b),
                    b,
                    0.5f * (b + c) };
        v4f bot = { 0.5f  * (a + d),
                    0.25f * (a + b + d + e),
                    0.5f  * (b + e),
                    0.25f * (b + c + e + f) };

        const size_t orow = (size_t)(2 * (r0 + ri));
        float* o0 = outP + orow * WO + 4 * cg;
        // Streaming writes: non-temporal so the 201MB output doesn't evict the
        // L2-resident 50MB input. Lowers to global_store_b128 ... th:TH_*_NT.
        __builtin_nontemporal_store(top, (v4f*)(o0));
        __builtin_nontemporal_store(bot, (v4f*)(o0 + WO));
    }
}

extern "C" void kernel_launch(void* const* d_in, const int* in_sizes, int n_in,
                              void* d_out, int out_size, void* d_ws, size_t ws_size,
                              hipStream_t stream) {
    const float* x   = (const float*)d_in[0];   // (16,3,512,512) f32
    float*       out = (float*)d_out;           // (16,3,1024,1024) f32
    // d_in[1] / d_in[2] are target_height/width (1024/1024) -> baked in.

    dim3 grid(1, HI / TIR, PLANES);  // (1, 64, 48) = 3072 workgroups
    dim3 block(256, 1, 1);           // 8 wave32 waves
    resize2x_async_kernel<<<grid, block, 0, stream>>>(x, out);
}